// GraphAttentionLayer_30863634989332
// MI455X (gfx1250) — compile-verified
//
#include <hip/hip_runtime.h>
#include <math.h>

typedef float v2f __attribute__((ext_vector_type(2)));
typedef float v8f __attribute__((ext_vector_type(8)));

#define GAT_ALPHA 0.2f
#define MASK_VAL -9000000000000000.0f

constexpr int BATCH = 64;   // B*N = 8*8
constexpr int NODES = 512;
constexpr int FEAT  = 256;  // IF == OF

// LDS row pads: stride % 64 == 4 -> WMMA A-fragment reads are bank-conflict free
constexpr int PAD_H   = 260; // 16x256 strip
constexpr int PAD_ATT = 516; // 32x512 attention rows

// Async global->LDS staging (gfx1250 path), guarded so host pass / older
// toolchains fall back to plain LDS stores.
#if defined(__AMDGCN__) && __has_builtin(__builtin_amdgcn_global_load_async_to_lds_b128)
#define USE_ASYNC_LDS 1
typedef int v4i_vs __attribute__((vector_size(4 * sizeof(int))));
#endif

// ---------------------------------------------------------------------------
// Kernel 0: interleave W (256x256) into K-pair layout:
//   Wint[(k>>1)*2F + 2n + (k&1)] = W[k*F + n]
// so a WMMA B-fragment (B[kb][n], B[kb+1][n]) is one contiguous b64 load.
// ---------------------------------------------------------------------------
__global__ __launch_bounds__(256) void gat_interleave_w(
    const float* __restrict__ W, float* __restrict__ Wint)
{
  const int k = blockIdx.x;     // 0..255
  const int n = threadIdx.x;    // 0..255
  Wint[(k >> 1) * (2 * FEAT) + 2 * n + (k & 1)] = W[k * FEAT + n];
}

// ---------------------------------------------------------------------------
// Kernel 1: Wh[b] = h[b] @ W  (fp32 WMMA), stored K-pair interleaved for
// kernel 2's B-fragments; src/dst projections fused from the live D tiles.
// grid.x = 64*32 ; block = 256 (8 waves); block: 16 rows x 256 cols
// ---------------------------------------------------------------------------
__global__ __launch_bounds__(256) void gat_gemm_wh(
    const float* __restrict__ h, const float* __restrict__ Wint,
    const float* __restrict__ a,
    float* __restrict__ WhB,    // (64, 256, 256, 2) interleaved
    float* __restrict__ srcv, float* __restrict__ dstv)
{
  __shared__ float sh[16 * PAD_H];
  __shared__ float psrc[8 * 16];
  __shared__ float pdst[8 * 16];

  const int b  = blockIdx.x >> 5;
  const int i0 = (blockIdx.x & 31) << 4;
  const int t  = threadIdx.x;

  // Stage 16x256 strip of h into LDS (async-to-LDS when available)
  {
    const float* s = h + (size_t)(b * NODES + i0) * FEAT;
    const int r = t >> 4;
    const int c = (t & 15) << 4;
#ifdef USE_ASYNC_LDS
#pragma unroll
    for (int q = 0; q < 4; ++q) {
      __builtin_amdgcn_global_load_async_to_lds_b128(
          (v4i_vs*)(s + r * FEAT + c + 4 * q),
          (v4i_vs*)&sh[r * PAD_H + c + 4 * q], 0, 0);
    }
    asm volatile("s_wait_asynccnt 0x0" ::: "memory");
#else
    const float4* g = (const float4*)(s + r * FEAT + c);
    float4 x0 = g[0], x1 = g[1], x2 = g[2], x3 = g[3];
    float* d = &sh[r * PAD_H + c];
    *(float4*)(d + 0) = x0;  *(float4*)(d + 4)  = x1;
    *(float4*)(d + 8) = x2;  *(float4*)(d + 12) = x3;
#endif
  }
  __syncthreads();

  const int lane = t & 31;
  const int wave = t >> 5;
  const int row  = lane & 15;          // A-fragment: lane -> M ; B/D: lane -> N
  const int kh   = lane >> 4;          // lane half -> K offset
  const int n0   = (wave << 5) + row;  // wave owns cols [32w, 32w+32)

  v8f acc0 = {}; v8f acc1 = {};
  for (int k = 0; k < FEAT; k += 4) {
    const int kb = k + (kh << 1);                       // even
    v2f afr; afr.x = sh[row * PAD_H + kb]; afr.y = sh[row * PAD_H + kb + 1];
    v2f b0 = *(const v2f*)(Wint + (kb >> 1) * (2 * FEAT) + 2 * n0);
    v2f b1 = *(const v2f*)(Wint + (kb >> 1) * (2 * FEAT) + 2 * (n0 + 16));
    acc0 = __builtin_amdgcn_wmma_f32_16x16x4_f32(false, afr, false, b0, (short)0, acc0, false, false);
    acc1 = __builtin_amdgcn_wmma_f32_16x16x4_f32(false, afr, false, b1, (short)0, acc1, false, false);
  }

  // ---- store Wh in K-pair interleaved layout (node index is K of GEMM 2) ----
  float* ob = WhB + (size_t)b * NODES * FEAT + (size_t)(i0 >> 1) * (2 * FEAT);
#pragma unroll
  for (int v = 0; v < 8; ++v) {
    const int m = v + (kh << 3);       // local row 0..15
    const int base = (m >> 1) * (2 * FEAT) + (m & 1);
    ob[base + 2 * n0]        = acc0[v];
    ob[base + 2 * (n0 + 16)] = acc1[v];
  }

  // ---- fused src/dst projections: src[i] = Wh[i,:].a1 ; dst[i] = Wh[i,:].a2 ----
  const float a1lo = a[n0], a1hi = a[n0 + 16];
  const float a2lo = a[FEAT + n0], a2hi = a[FEAT + n0 + 16];
#pragma unroll
  for (int v = 0; v < 8; ++v) {
    float c1 = fmaf(acc0[v], a1lo, acc1[v] * a1hi);
    float c2 = fmaf(acc0[v], a2lo, acc1[v] * a2hi);
#pragma unroll
    for (int off = 1; off < 16; off <<= 1) {   // reduce within each 16-lane half
      c1 += __shfl_xor(c1, off, 32);
      c2 += __shfl_xor(c2, off, 32);
    }
    if ((lane & 15) == 0) {                    // lanes 0 and 16
      const int m = v + (kh << 3);
      psrc[wave * 16 + m] = c1;
      pdst[wave * 16 + m] = c2;
    }
  }
  __syncthreads();
  if (t < 16) {
    float s = 0.f;
#pragma unroll
    for (int w = 0; w < 8; ++w) s += psrc[w * 16 + t];
    srcv[b * NODES + i0 + t] = s;
  } else if (t < 32) {
    const int r2 = t - 16;
    float s = 0.f;
#pragma unroll
    for (int w = 0; w < 8; ++w) s += pdst[w * 16 + r2];
    dstv[b * NODES + i0 + r2] = s;
  }
}

// ---------------------------------------------------------------------------
// Kernel 2: fused masked-softmax attention + h_prime GEMM + ELU.
// Block handles 32 attention rows so each B-fragment feeds two row-tiles
// (halves L2 traffic, doubles WMMA per load).
// grid.x = 64*16 ; block = 256 (8 waves)
// ---------------------------------------------------------------------------
__global__ __launch_bounds__(256) void gat_attn_out(
    const float* __restrict__ WhB, const int* __restrict__ adj,
    const float* __restrict__ srcv, const float* __restrict__ dstv,
    float* __restrict__ out)
{
  __shared__ float att[32 * PAD_ATT];   // 32 x 512 (padded) ~ 66 KB
  const int b    = blockIdx.x >> 4;
  const int i0   = (blockIdx.x & 15) << 5;
  const int t    = threadIdx.x;
  const int lane = t & 31;
  const int wave = t >> 5;

  // --- masked leaky-relu logits + row softmax (4 rows per wave) ---
#pragma unroll
  for (int rr = 0; rr < 4; ++rr) {
    const int lr = (wave << 2) + rr;          // local row 0..31
    const int i  = i0 + lr;
    const float si = srcv[b * NODES + i];
    const int*   arow = adj + (size_t)i * NODES;
    const float* drow = dstv + b * NODES;

    float m = -INFINITY;
#pragma unroll
    for (int jt = 0; jt < NODES / 32; ++jt) {
      const int j = lane + (jt << 5);
      float e;
      if (arow[j] > 0) {
        const float x = si + drow[j];
        e = x > 0.f ? x : GAT_ALPHA * x;
      } else {
        e = MASK_VAL;
      }
      att[lr * PAD_ATT + j] = e;
      m = fmaxf(m, e);
    }
#pragma unroll
    for (int off = 16; off > 0; off >>= 1) m = fmaxf(m, __shfl_xor(m, off, 32));

    float sum = 0.f;
#pragma unroll
    for (int jt = 0; jt < NODES / 32; ++jt) {
      const int j = lane + (jt << 5);
      const float p = __expf(att[lr * PAD_ATT + j] - m);
      att[lr * PAD_ATT + j] = p;
      sum += p;
    }
#pragma unroll
    for (int off = 16; off > 0; off >>= 1) sum += __shfl_xor(sum, off, 32);

    const float inv = 1.f / sum;
#pragma unroll
    for (int jt = 0; jt < NODES / 32; ++jt) {
      const int j = lane + (jt << 5);
      att[lr * PAD_ATT + j] *= inv;
    }
  }
  __syncthreads();

  // --- h_prime(32x256) = att(32x512) @ Wh_b(512x256), then ELU ---
  const float* Bb = WhB + (size_t)b * NODES * FEAT;   // K-pair interleaved
  const int row = lane & 15;
  const int kh  = lane >> 4;
  const int n0  = (wave << 5) + row;

  v8f accP0 = {}; v8f accP1 = {};   // att rows 0..15
  v8f accQ0 = {}; v8f accQ1 = {};   // att rows 16..31
  for (int k = 0; k < NODES; k += 4) {
    const int kb = k + (kh << 1);                     // even
    v2f aP; aP.x = att[row * PAD_ATT + kb];        aP.y = att[row * PAD_ATT + kb + 1];
    v2f aQ; aQ.x = att[(16 + row) * PAD_ATT + kb]; aQ.y = att[(16 + row) * PAD_ATT + kb + 1];
    v2f b0 = *(const v2f*)(Bb + (kb >> 1) * (2 * FEAT) + 2 * n0);
    v2f b1 = *(const v2f*)(Bb + (kb >> 1) * (2 * FEAT) + 2 * (n0 + 16));
    accP0 = __builtin_amdgcn_wmma_f32_16x16x4_f32(false, aP, false, b0, (short)0, accP0, false, false);
    accP1 = __builtin_amdgcn_wmma_f32_16x16x4_f32(false, aP, false, b1, (short)0, accP1, false, false);
    accQ0 = __builtin_amdgcn_wmma_f32_16x16x4_f32(false, aQ, false, b0, (short)0, accQ0, false, false);
    accQ1 = __builtin_amdgcn_wmma_f32_16x16x4_f32(false, aQ, false, b1, (short)0, accQ1, false, false);
  }

  float* o = out + (size_t)(b * NODES + i0) * FEAT;
#pragma unroll
  for (int v = 0; v < 8; ++v) {
    const int m = v + (kh << 3);
    const float p0 = accP0[v], p1 = accP1[v];
    const float q0 = accQ0[v], q1 = accQ1[v];
    o[m * FEAT + n0]             = (p0 > 0.f) ? p0 : expm1f(p0);   // elu, alpha=1
    o[m * FEAT + n0 + 16]        = (p1 > 0.f) ? p1 : expm1f(p1);
    o[(16 + m) * FEAT + n0]      = (q0 > 0.f) ? q0 : expm1f(q0);
    o[(16 + m) * FEAT + n0 + 16] = (q1 > 0.f) ? q1 : expm1f(q1);
  }
}

// ---------------------------------------------------------------------------
extern "C" void kernel_launch(void* const* d_in, const int* in_sizes, int n_in,
                              void* d_out, int out_size, void* d_ws, size_t ws_size,
                              hipStream_t stream) {
  (void)in_sizes; (void)n_in; (void)out_size; (void)ws_size;
  const float* h   = (const float*)d_in[0];   // (8,8,512,256) f32
  const int*   adj = (const int*)  d_in[1];   // (512,512) i32
  const float* W   = (const float*)d_in[2];   // (256,256) f32
  const float* a   = (const float*)d_in[3];   // (512,1) f32
  float* out = (float*)d_out;                 // (8,8,512,256) f32

  float* Wint = (float*)d_ws;                                  // 256*256
  float* WhB  = Wint + (size_t)FEAT * FEAT;                    // 64*512*256 (interleaved)
  float* srcv = WhB + (size_t)BATCH * NODES * FEAT;            // 64*512
  float* dstv = srcv + (size_t)BATCH * NODES;                  // 64*512

  hipLaunchKernelGGL(gat_interleave_w, dim3(FEAT),      dim3(FEAT), 0, stream, W, Wint);
  hipLaunchKernelGGL(gat_gemm_wh,      dim3(BATCH * 32), dim3(256), 0, stream,
                     h, Wint, a, WhB, srcv, dstv);
  hipLaunchKernelGGL(gat_attn_out,     dim3(BATCH * 16), dim3(256), 0, stream,
                     WhB, adj, srcv, dstv, out);
}